// GraphEncoder_Attn_7825430413998
// MI455X (gfx1250) — compile-verified
//
#include <hip/hip_runtime.h>
#include <hip/hip_bf16.h>

// ---------------- problem constants (fixed by the reference) ----------------
#define B_    2
#define S_    512
#define N_    20000
#define D_    384
#define H_    3
#define C_    128
#define E_    320000
#define DG_   128
#define F_    (H_ * C_)          // 384
#define BN_   (B_ * N_)          // 40000
#define ETOT_ (B_ * E_ + BN_)    // 680000
#define NEG_SLOPE 0.2f
#define KSPLIT_ 20               // K-split for the S x N x F projection GEMM
#define KCH_    (N_ / KSPLIT_)   // 1000
#define NB_   4                  // 16x16 tiles per wave along N (16x64 strip)

typedef float v2f __attribute__((ext_vector_type(2)));
typedef float v8f __attribute__((ext_vector_type(8)));

// ---------------- WMMA 16x64 strip helpers (V_WMMA_F32_16X16X4_F32) ---------
// One wave computes a 16x64 output strip: the A fragment (16x4) is loaded once
// per k-step and reused by 4 WMMAs against 4 adjacent B fragments. This
// quadruples FLOPs per A fetch vs a single-tile wave.
//  A  : lanes 0-15 hold M=lane, v0/v1 = K{0,1}; lanes 16-31 = K{2,3}
//  B  : v0 lanes0-15 = row K=0 (N=lane), lanes16-31 = row K=2; v1 = K1/K3
//  C/D: lane l<16 -> N=l, M=vgpr j; lanes 16-31 -> M=j+8
__device__ __forceinline__ void wmma_strip_nb(const float* __restrict__ A, long lda,
                                              const float* __restrict__ Bm, long ldb,
                                              int K, v8f* acc) {
  const int lane = threadIdx.x & 31;
  const int half = lane >> 4;
  const int l15  = lane & 15;
  const float* a = A + (long)l15 * lda + half * 2;
  const float* b = Bm + (long)(half * 2) * ldb + l15;
  for (int k = 0; k < K; k += 4) {
    v2f av; av[0] = a[0]; av[1] = a[1];
#pragma unroll
    for (int i = 0; i < NB_; i++) {
      v2f bv; bv[0] = b[16 * i]; bv[1] = b[ldb + 16 * i];
      acc[i] = __builtin_amdgcn_wmma_f32_16x16x4_f32(false, av, false, bv,
                                                     (short)0, acc[i], false, false);
    }
    a += 4;
    b += 4 * ldb;
  }
}

// Same, but B is stored transposed (row n contiguous over k, stride ldbt).
__device__ __forceinline__ void wmma_strip_bt(const float* __restrict__ A, long lda,
                                              const float* __restrict__ Bt, long ldbt,
                                              int K, v8f* acc) {
  const int lane = threadIdx.x & 31;
  const int half = lane >> 4;
  const int l15  = lane & 15;
  const float* a = A + (long)l15 * lda + half * 2;
  const float* b = Bt + (long)l15 * ldbt + half * 2;
  for (int k = 0; k < K; k += 4) {
    v2f av; av[0] = a[0]; av[1] = a[1];
#pragma unroll
    for (int i = 0; i < NB_; i++) {
      v2f bv; bv[0] = b[(long)16 * i * ldbt]; bv[1] = b[(long)16 * i * ldbt + 1];
      acc[i] = __builtin_amdgcn_wmma_f32_16x16x4_f32(false, av, false, bv,
                                                     (short)0, acc[i], false, false);
    }
    a += 4;
    b += 4;
  }
}

// ---------------- GEMM1: x = Wq @ x_enc[b] + bq + mask*Wm + bm --------------
// out x[BN, 384]; M=N_ K=S_ Ncol=D_  (strips: 1250 x 6 per batch)
__global__ void k_gemm1(const float* __restrict__ x_enc, const float* __restrict__ mask,
                        const float* __restrict__ Wq, const float* __restrict__ bq,
                        const float* __restrict__ Wm, const float* __restrict__ bm,
                        float* __restrict__ x) {
  const int TM = N_ / 16, TN = D_ / (16 * NB_);  // 1250, 6
  int wave = (blockIdx.x * blockDim.x + threadIdx.x) >> 5;
  if (wave >= B_ * TM * TN) return;
  int b  = wave / (TM * TN);
  int r  = wave % (TM * TN);
  int tm = r / TN, tn = r % TN;

  v8f acc[NB_] = {};
  wmma_strip_nb(Wq + (long)tm * 16 * S_, S_,
                x_enc + (long)b * S_ * D_ + tn * 16 * NB_, D_, S_, acc);

  const int lane = threadIdx.x & 31, half = lane >> 4, l15 = lane & 15;
  float wmc[NB_], bmc[NB_];
#pragma unroll
  for (int i = 0; i < NB_; i++) {
    int col = tn * 16 * NB_ + i * 16 + l15;
    wmc[i] = Wm[col];
    bmc[i] = bm[col];
  }
#pragma unroll
  for (int j = 0; j < 8; j++) {
    int row = tm * 16 + half * 8 + j;
    float bqr = bq[row];
    float mk  = mask[b * N_ + row];
    float* xr = x + ((long)(b * N_ + row)) * F_ + tn * 16 * NB_ + l15;
#pragma unroll
    for (int i = 0; i < NB_; i++)
      xr[i * 16] = acc[i][j] + bqr + mk * wmc[i] + bmc[i];
  }
}

// ---------------- GEMM2: xp = x @ Wlin^T ------------------------------------
// M=BN_ K=D_ Ncol=F_  (strips: 2500 x 6)
__global__ void k_gemm2(const float* __restrict__ x, const float* __restrict__ Wlin,
                        float* __restrict__ xp) {
  const int TM = BN_ / 16, TN = F_ / (16 * NB_);  // 2500, 6
  int wave = (blockIdx.x * blockDim.x + threadIdx.x) >> 5;
  if (wave >= TM * TN) return;
  int tm = wave / TN, tn = wave % TN;

  v8f acc[NB_] = {};
  wmma_strip_bt(x + (long)tm * 16 * D_, D_,
                Wlin + (long)tn * 16 * NB_ * D_, D_, D_, acc);

  const int lane = threadIdx.x & 31, half = lane >> 4, l15 = lane & 15;
#pragma unroll
  for (int j = 0; j < 8; j++) {
    float* p = xp + ((long)tm * 16 + half * 8 + j) * F_ + tn * 16 * NB_ + l15;
#pragma unroll
    for (int i = 0; i < NB_; i++) p[i * 16] = acc[i][j];
  }
}

// ---------------- per-node attention dots -----------------------------------
__global__ void k_node_att(const float* __restrict__ xp,
                           const float* __restrict__ att_src,
                           const float* __restrict__ att_dst,
                           float* __restrict__ a_src, float* __restrict__ a_dst) {
  int wave = (blockIdx.x * blockDim.x + threadIdx.x) >> 5;
  if (wave >= BN_ * H_) return;
  int i = wave / H_, h = wave % H_;
  int lane = threadIdx.x & 31;
  const float* xr = xp + (long)i * F_ + h * C_;
  float s0 = 0.f, s1 = 0.f;
#pragma unroll
  for (int k = 0; k < C_; k += 32) {
    float v = xr[k + lane];
    s0 += v * att_src[h * C_ + k + lane];
    s1 += v * att_dst[h * C_ + k + lane];
  }
#pragma unroll
  for (int m = 16; m >= 1; m >>= 1) {
    s0 += __shfl_xor(s0, m, 32);
    s1 += __shfl_xor(s1, m, 32);
  }
  if (lane == 0) { a_src[wave] = s0; a_dst[wave] = s1; }
}

// ---------------- edge helpers ----------------------------------------------
__device__ __forceinline__ void edge_nodes(int t, const int* __restrict__ ei,
                                           int& s, int& d) {
  if (t < B_ * E_) {
    int b = t / E_, j = t - b * E_;
    s = ei[j] + b * N_;
    d = ei[E_ + j] + b * N_;
  } else {
    s = d = t - B_ * E_;  // self loop
  }
}

__device__ __forceinline__ void atomicMaxF32(float* addr, float val) {
  unsigned int* ua = (unsigned int*)addr;
  unsigned int old = *ua;
  while (__uint_as_float(old) < val) {
    unsigned int assumed = old;
    old = atomicCAS(ua, assumed, __float_as_uint(val));
    if (old == assumed) break;
  }
}

__device__ __forceinline__ float leaky(float v) {
  return v > 0.f ? v : NEG_SLOPE * v;
}

__global__ void k_fill(float* __restrict__ p, long n, float v) {
  long t = (long)blockIdx.x * blockDim.x + threadIdx.x;
  if (t < n) p[t] = v;
}

__global__ void k_proj_init(const float* __restrict__ bproj, float* __restrict__ proj) {
  long t = (long)blockIdx.x * blockDim.x + threadIdx.x;
  if (t >= (long)B_ * S_ * F_) return;
  int s = (int)((t / F_) % S_);
  proj[t] = bproj[s];
}

// Pass 1: segment max over destination nodes
__global__ void k_edge_max(const float* __restrict__ a_src, const float* __restrict__ a_dst,
                           const int* __restrict__ ei, float* __restrict__ emax) {
  int t = blockIdx.x * blockDim.x + threadIdx.x;
  if (t >= ETOT_) return;
  int s, d;
  edge_nodes(t, ei, s, d);
#pragma unroll
  for (int h = 0; h < H_; h++) {
    float e = leaky(a_src[s * H_ + h] + a_dst[d * H_ + h]);
    atomicMaxF32(&emax[d * H_ + h], e);
  }
}

// Pass 2: ee = exp(e - emax[dst]); denom[dst] += ee
__global__ void k_edge_exp(const float* __restrict__ a_src, const float* __restrict__ a_dst,
                           const int* __restrict__ ei, const float* __restrict__ emax,
                           float* __restrict__ ee, float* __restrict__ denom) {
  int t = blockIdx.x * blockDim.x + threadIdx.x;
  if (t >= ETOT_) return;
  int s, d;
  edge_nodes(t, ei, s, d);
#pragma unroll
  for (int h = 0; h < H_; h++) {
    float e  = leaky(a_src[s * H_ + h] + a_dst[d * H_ + h]);
    float ex = __expf(e - emax[d * H_ + h]);
    ee[(long)t * H_ + h] = ex;
    atomicAdd(&denom[d * H_ + h], ex);
  }
}

// Pass 3: accum[dst, f] += ee * xp[src, f]  (one wave per edge)
__global__ void k_edge_scatter(const float* __restrict__ xp, const float* __restrict__ ee,
                               const int* __restrict__ ei, float* __restrict__ accum) {
  int wave = (blockIdx.x * blockDim.x + threadIdx.x) >> 5;
  if (wave >= ETOT_) return;
  int lane = threadIdx.x & 31;
  int s, d;
  edge_nodes(wave, ei, s, d);
  const float e0 = ee[(long)wave * H_ + 0];
  const float e1 = ee[(long)wave * H_ + 1];
  const float e2 = ee[(long)wave * H_ + 2];
  const float* xr = xp + (long)s * F_;
  float* ar = accum + (long)d * F_;
#pragma unroll
  for (int f0 = 0; f0 < F_; f0 += 32) {
    int f = f0 + lane;
    float w = (f < C_) ? e0 : ((f < 2 * C_) ? e1 : e2);
    atomicAdd(&ar[f], w * xr[f]);
  }
}

// Pass 4: gat = accum / denom + bias (in place)
__global__ void k_gat_final(float* __restrict__ accum, const float* __restrict__ denom,
                            const float* __restrict__ bias_gat) {
  long t = (long)blockIdx.x * blockDim.x + threadIdx.x;
  if (t >= (long)BN_ * F_) return;
  int i = (int)(t / F_), f = (int)(t % F_);
  int h = f >> 7;  // f / C_
  accum[t] = accum[t] / denom[i * H_ + h] + bias_gat[f];
}

// ---------------- GEMM3: proj[b] += Wproj @ gat[b]  (K split, atomic) -------
// M=S_ K=N_ Ncol=F_ (strips 32 x 6 per batch per k-chunk, 20 chunks)
__global__ void k_gemm3(const float* __restrict__ Wproj, const float* __restrict__ gat,
                        float* __restrict__ proj) {
  const int TM = S_ / 16, TN = F_ / (16 * NB_);  // 32, 6
  int wave = (blockIdx.x * blockDim.x + threadIdx.x) >> 5;
  if (wave >= B_ * KSPLIT_ * TM * TN) return;
  int tile = wave % (TM * TN);
  int bk   = wave / (TM * TN);
  int b    = bk % B_;
  int kc   = bk / B_;
  int tm = tile / TN, tn = tile % TN;

  v8f acc[NB_] = {};
  wmma_strip_nb(Wproj + (long)tm * 16 * N_ + (long)kc * KCH_, N_,
                gat + (long)b * N_ * F_ + (long)kc * KCH_ * F_ + tn * 16 * NB_, F_,
                KCH_, acc);

  const int lane = threadIdx.x & 31, half = lane >> 4, l15 = lane & 15;
#pragma unroll
  for (int j = 0; j < 8; j++) {
    int row = tm * 16 + half * 8 + j;
    float* p = proj + ((long)b * S_ + row) * F_ + tn * 16 * NB_ + l15;
#pragma unroll
    for (int i = 0; i < NB_; i++) atomicAdd(&p[i * 16], acc[i][j]);
  }
}

// ---------------- GEMM4: out = proj @ Wtemp^T + btemp -----------------------
// M=B_*S_ K=F_ Ncol=DG_ (strips 64 x 2)
__global__ void k_gemm4(const float* __restrict__ proj, const float* __restrict__ Wtemp,
                        const float* __restrict__ btemp, float* __restrict__ out) {
  const int TM = (B_ * S_) / 16, TN = DG_ / (16 * NB_);  // 64, 2
  int wave = (blockIdx.x * blockDim.x + threadIdx.x) >> 5;
  if (wave >= TM * TN) return;
  int tm = wave / TN, tn = wave % TN;

  v8f acc[NB_] = {};
  wmma_strip_bt(proj + (long)tm * 16 * F_, F_,
                Wtemp + (long)tn * 16 * NB_ * F_, F_, F_, acc);

  const int lane = threadIdx.x & 31, half = lane >> 4, l15 = lane & 15;
  float bt[NB_];
#pragma unroll
  for (int i = 0; i < NB_; i++) bt[i] = btemp[tn * 16 * NB_ + i * 16 + l15];
#pragma unroll
  for (int j = 0; j < 8; j++) {
    float* p = out + ((long)tm * 16 + half * 8 + j) * DG_ + tn * 16 * NB_ + l15;
#pragma unroll
    for (int i = 0; i < NB_; i++) p[i * 16] = acc[i][j] + bt[i];
  }
}

// ---------------- host-side orchestration -----------------------------------
extern "C" void kernel_launch(void* const* d_in, const int* in_sizes, int n_in,
                              void* d_out, int out_size, void* d_ws, size_t ws_size,
                              hipStream_t stream) {
  (void)in_sizes; (void)n_in; (void)out_size; (void)ws_size;
  const float* x_enc    = (const float*)d_in[0];
  const float* mask     = (const float*)d_in[1];
  const int*   ei       = (const int*)  d_in[2];
  const float* Wq       = (const float*)d_in[3];
  const float* bq       = (const float*)d_in[4];
  const float* Wm       = (const float*)d_in[5];
  const float* bm       = (const float*)d_in[6];
  const float* Wlin     = (const float*)d_in[7];
  const float* att_src  = (const float*)d_in[8];
  const float* att_dst  = (const float*)d_in[9];
  const float* bias_gat = (const float*)d_in[10];
  const float* Wproj    = (const float*)d_in[11];
  const float* bproj    = (const float*)d_in[12];
  const float* Wtemp    = (const float*)d_in[13];
  const float* btemp    = (const float*)d_in[14];
  float* out = (float*)d_out;

  // workspace layout (floats)
  float* ws     = (float*)d_ws;
  float* x      = ws;                         // BN_*F_   (reused as accum/gat)
  float* xp     = x  + (long)BN_ * F_;        // BN_*F_
  float* a_src  = xp + (long)BN_ * F_;        // BN_*H_
  float* a_dst  = a_src + (long)BN_ * H_;     // BN_*H_
  float* emax   = a_dst + (long)BN_ * H_;     // BN_*H_
  float* denom  = emax  + (long)BN_ * H_;     // BN_*H_
  float* ee     = denom + (long)BN_ * H_;     // ETOT_*H_
  float* proj   = ee + (long)ETOT_ * H_;      // B_*S_*F_

  const int TB = 256;  // 8 waves/block

  // 1. x = Wq @ x_enc + bq + mask-embed      (15000 waves, 16x64 strips)
  k_gemm1<<<(B_ * (N_ / 16) * (D_ / 64)) / 8, TB, 0, stream>>>(
      x_enc, mask, Wq, bq, Wm, bm, x);

  // 2. xp = x @ Wlin^T                       (15000 waves)
  k_gemm2<<<((BN_ / 16) * (F_ / 64)) / 8, TB, 0, stream>>>(x, Wlin, xp);

  // 3. per-node attention scalars            (120000 waves)
  k_node_att<<<(BN_ * H_) / 8, TB, 0, stream>>>(xp, att_src, att_dst, a_src, a_dst);

  // 4. init scatter state (x buffer reused as accumulator)
  k_fill<<<((long)BN_ * F_ + TB - 1) / TB, TB, 0, stream>>>(x, (long)BN_ * F_, 0.f);
  k_fill<<<(BN_ * H_ + TB - 1) / TB, TB, 0, stream>>>(emax, (long)BN_ * H_, -1e30f);
  k_fill<<<(BN_ * H_ + TB - 1) / TB, TB, 0, stream>>>(denom, (long)BN_ * H_, 0.f);
  k_proj_init<<<((long)B_ * S_ * F_ + TB - 1) / TB, TB, 0, stream>>>(bproj, proj);

  // 5-7. segment softmax + scatter
  k_edge_max<<<(ETOT_ + TB - 1) / TB, TB, 0, stream>>>(a_src, a_dst, ei, emax);
  k_edge_exp<<<(ETOT_ + TB - 1) / TB, TB, 0, stream>>>(a_src, a_dst, ei, emax, ee, denom);
  k_edge_scatter<<<ETOT_ / 8, TB, 0, stream>>>(xp, ee, ei, x);

  // 8. normalize + bias (in place -> gat)
  k_gat_final<<<((long)BN_ * F_) / TB, TB, 0, stream>>>(x, denom, bias_gat);

  // 9. proj = Wproj @ gat + bproj            (7680 waves, K-split atomics)
  k_gemm3<<<(B_ * KSPLIT_ * (S_ / 16) * (F_ / 64)) / 8, TB, 0, stream>>>(Wproj, x, proj);

  // 10. out = proj @ Wtemp^T + btemp         (128 waves)
  k_gemm4<<<(((B_ * S_) / 16) * (DG_ / 64)) / 8, TB, 0, stream>>>(proj, Wtemp, btemp, out);
}